// MultiHeadAttentionLayer_17377437679643
// MI455X (gfx1250) — compile-verified
//
#include <hip/hip_runtime.h>
#include <hip/hip_fp16.h>
#include <stdint.h>

#define B_ 2
#define N_ 512
#define IN_DIM_ 128
#define H_ 8
#define D_ 16
#define HD_ 128

typedef __attribute__((ext_vector_type(2))) float v2f;
typedef __attribute__((ext_vector_type(8))) float v8f;
typedef __attribute__((ext_vector_type(4))) int   v4i;
typedef __attribute__((address_space(1))) v4i* gv4i_p;   // global (AS1) 4xi32
typedef __attribute__((address_space(3))) v4i* lv4i_p;   // LDS    (AS3) 4xi32

#if defined(__HIP_DEVICE_COMPILE__) && defined(__has_builtin)
#if __has_builtin(__builtin_amdgcn_global_load_async_to_lds_b128) && \
    __has_builtin(__builtin_amdgcn_s_wait_asynccnt)
#define USE_ASYNC_LDS 1
#endif
#endif

// ---------------------------------------------------------------------------
// Stage A: Q/K/V projections.  One block per (b,i) row; thread t = channel k.
// Q,K stored head-major (B*H, N, D); V stored (B, N, HD).  K pre-scaled 1/sqrt(D).
// ---------------------------------------------------------------------------
__global__ __launch_bounds__(IN_DIM_) void proj_kernel(
    const float* __restrict__ h,
    const float* __restrict__ Wq, const float* __restrict__ Wk,
    const float* __restrict__ Wv,
    float* __restrict__ Qw, float* __restrict__ Kw, float* __restrict__ Vw) {
  const int row = blockIdx.x;          // b*N + i
  const int t = threadIdx.x;           // output channel k (0..127)
  __shared__ float hs[IN_DIM_];
  hs[t] = h[(size_t)row * IN_DIM_ + t];
  __syncthreads();
  float q = 0.f, k = 0.f, v = 0.f;
#pragma unroll 8
  for (int c = 0; c < IN_DIM_; ++c) {
    const float hc = hs[c];
    q = fmaf(hc, Wq[c * HD_ + t], q);
    k = fmaf(hc, Wk[c * HD_ + t], k);
    v = fmaf(hc, Wv[c * HD_ + t], v);
  }
  const int b = row / N_, i = row % N_;
  const int hh = t >> 4, d = t & 15;
  Qw[(((size_t)b * H_ + hh) * N_ + i) * D_ + d] = q;
  Kw[(((size_t)b * H_ + hh) * N_ + i) * D_ + d] = k * 0.25f;  // D^-0.5
  Vw[(size_t)row * HD_ + t] = v;
}

// ---------------------------------------------------------------------------
// Stage B: scores[bh][i][j] = sum_d Q[bh][i][d] * K[bh][j][d]
// One wave32 per 16x16 tile; 4 chained V_WMMA_F32_16X16X4_F32 (K=4 each).
// ---------------------------------------------------------------------------
__global__ __launch_bounds__(32) void qk_wmma_kernel(
    const float* __restrict__ Qw, const float* __restrict__ Kw,
    float* __restrict__ qk) {
  const int tiles = N_ / 16;                      // 32
  const int tj = blockIdx.x % tiles;
  const int ti = (blockIdx.x / tiles) % tiles;
  const int bh = blockIdx.x / (tiles * tiles);    // 0..B*H-1
  const float* __restrict__ Qbase = Qw + (size_t)bh * N_ * D_;
  const float* __restrict__ Kbase = Kw + (size_t)bh * N_ * D_;

  const int lane  = threadIdx.x;
  const int mrow  = ti * 16 + (lane & 15);
  const int ncol  = tj * 16 + (lane & 15);
  const int khalf = (lane >> 4) << 1;             // 0 or 2

  v8f c = {};
#if __has_builtin(__builtin_amdgcn_wmma_f32_16x16x4_f32)
#pragma unroll
  for (int kb = 0; kb < D_; kb += 4) {
    const int kd = kb + khalf;                    // even -> 8B aligned
    v2f a = *(const v2f*)&Qbase[mrow * D_ + kd];
    v2f b = *(const v2f*)&Kbase[ncol * D_ + kd];
    c = __builtin_amdgcn_wmma_f32_16x16x4_f32(
        false, a, false, b, (short)0, c, false, false);
  }
#else
  for (int r = 0; r < 8; ++r) {
    const int mr = ti * 16 + r + 8 * (lane >> 4);
    float acc = 0.f;
    for (int d = 0; d < D_; ++d)
      acc = fmaf(Qbase[mr * D_ + d], Kbase[(tj * 16 + (lane & 15)) * D_ + d], acc);
    c[r] = acc;
  }
#endif
  const int rbase = ti * 16 + 8 * (lane >> 4);
  const int col   = tj * 16 + (lane & 15);
#pragma unroll
  for (int r = 0; r < 8; ++r)
    qk[((size_t)bh * N_ + (rbase + r)) * N_ + col] = c[r];
}

// ---------------------------------------------------------------------------
// Stage C: fused bias + softmax(j) + weighted sum; single coalesced pass over
// e_att / e_val (online softmax).  One block per (b,i) row:
//   256 threads = 8 j-groups x 32 lanes; each lane owns 4 consecutive channels
//   and streams with global_load_b128 (512B per wave per stream per j).
// Mask is folded into the LDS-staged qk scores (masked j -> -3e38 -> exp==0).
// ---------------------------------------------------------------------------
#define QSTR (N_ + 4)   // padded head stride: conflict-free LDS banks
#define PF_ROWS 8       // prefetch distance: 8 rows * 512B = 4KB ahead

#define ONLINE_UPD(C)                                   \
  {                                                     \
    const float s  = qj + a.C;                          \
    const float mn = fmaxf(m4.C, s);                    \
    const float cr = __expf(m4.C - mn);                 \
    const float p  = __expf(s - mn);                    \
    Z4.C = Z4.C * cr + p;                               \
    A4.C = A4.C * cr + p * (v.C + e.C);                 \
    m4.C = mn;                                          \
  }

#define MERGE_UPD(C)                                    \
  {                                                     \
    const float M  = fmaxf(m.C, mg.C);                  \
    const float c0 = __expf(m.C - M);                   \
    const float c1 = __expf(mg.C - M);                  \
    Z.C = Z.C * c0 + Zg.C * c1;                         \
    A.C = A.C * c0 + Ag.C * c1;                         \
    m.C = M;                                            \
  }

__global__ __launch_bounds__(256) void attn_stream_kernel(
    const float* __restrict__ e_att, const float* __restrict__ e_val,
    const uint8_t* __restrict__ mask, const float* __restrict__ Vw,
    const float* __restrict__ qk, float* __restrict__ out) {
  const int row = blockIdx.x;              // b*N + i
  const int b = row / N_, i = row % N_;
  const int t    = threadIdx.x;
  const int jg   = t >> 5;                 // j-group: 0..7
  const int lane = t & 31;                 // channel group: channels 4*lane..
  const int hh   = lane >> 2;              // head of this lane's 4 channels

  __shared__ float  qhm[H_ * QSTR];        // qk row + mask folded, padded stride
  __shared__ float4 smv[8][32], szv[8][32], sav[8][32];

  // ---- stage qk[b, :, i, :] into LDS (async-to-LDS path on CDNA5) ----
#ifdef USE_ASYNC_LDS
  for (int idx = t; idx < (H_ * N_) / 4; idx += 256) {
    const int e0 = idx << 2;
    const int h2 = e0 >> 9, j2 = e0 & (N_ - 1);
    const float* gsrc = qk + (((size_t)b * H_ + h2) * N_ + i) * N_ + j2;
    __builtin_amdgcn_global_load_async_to_lds_b128(
        (gv4i_p)(void*)gsrc, (lv4i_p)(void*)&qhm[h2 * QSTR + j2], 0, 0);
  }
  __builtin_amdgcn_s_wait_asynccnt(0);
  __syncthreads();
#else
  for (int idx = t; idx < H_ * N_; idx += 256) {
    const int h2 = idx >> 9, j2 = idx & (N_ - 1);
    qhm[h2 * QSTR + j2] = qk[(((size_t)b * H_ + h2) * N_ + i) * N_ + j2];
  }
  __syncthreads();
#endif
  // fold attention mask additively (masked j contributes exactly 0 via exp)
  for (int idx = t; idx < H_ * N_; idx += 256) {
    const int h2 = idx >> 9, j2 = idx & (N_ - 1);
    if (!mask[(size_t)row * N_ + j2]) qhm[h2 * QSTR + j2] = -3.0e38f;
  }
  __syncthreads();

  const float* __restrict__ qh  = &qhm[hh * QSTR];
  const float4* __restrict__ ea4 = (const float4*)(e_att + (size_t)row * N_ * HD_) + lane;
  const float4* __restrict__ ev4 = (const float4*)(e_val + (size_t)row * N_ * HD_) + lane;
  const float4* __restrict__ Vb4 = (const float4*)(Vw + (size_t)b * N_ * HD_) + lane;

  const int j0 = jg * (N_ / 8), j1 = j0 + (N_ / 8);
  float4 m4 = make_float4(-3.0e38f, -3.0e38f, -3.0e38f, -3.0e38f);
  float4 Z4 = make_float4(0.f, 0.f, 0.f, 0.f);
  float4 A4 = make_float4(0.f, 0.f, 0.f, 0.f);

  for (int j = j0; j < j1; ++j) {
    const int jp = (j + PF_ROWS < j1) ? j + PF_ROWS : j1 - 1;
    __builtin_prefetch(&ea4[(size_t)jp * (HD_ / 4)], 0, 1);  // global_prefetch_b8
    __builtin_prefetch(&ev4[(size_t)jp * (HD_ / 4)], 0, 1);
    const float4 a = ea4[(size_t)j * (HD_ / 4)];             // global_load_b128
    const float4 e = ev4[(size_t)j * (HD_ / 4)];
    const float4 v = Vb4[(size_t)j * (HD_ / 4)];
    const float qj = qh[j];
    ONLINE_UPD(x) ONLINE_UPD(y) ONLINE_UPD(z) ONLINE_UPD(w)
  }

  smv[jg][lane] = m4; szv[jg][lane] = Z4; sav[jg][lane] = A4;
  __syncthreads();

  if (t < 32) {
    float4 m = smv[0][t], Z = szv[0][t], A = sav[0][t];
#pragma unroll
    for (int g = 1; g < 8; ++g) {
      const float4 mg = smv[g][t], Zg = szv[g][t], Ag = sav[g][t];
      MERGE_UPD(x) MERGE_UPD(y) MERGE_UPD(z) MERGE_UPD(w)
    }
    float4 o;
    o.x = A.x / Z.x; o.y = A.y / Z.y; o.z = A.z / Z.z; o.w = A.w / Z.w;
    *(float4*)&out[(size_t)row * HD_ + t * 4] = o;
  }
}

// ---------------------------------------------------------------------------
extern "C" void kernel_launch(void* const* d_in, const int* in_sizes, int n_in,
                              void* d_out, int out_size, void* d_ws, size_t ws_size,
                              hipStream_t stream) {
  const float*   h     = (const float*)d_in[0];
  const float*   e_att = (const float*)d_in[1];
  const float*   e_val = (const float*)d_in[2];
  const uint8_t* mask  = (const uint8_t*)d_in[3];
  const float*   Wq    = (const float*)d_in[4];
  const float*   Wk    = (const float*)d_in[5];
  const float*   Wv    = (const float*)d_in[6];
  float* out = (float*)d_out;

  float* ws = (float*)d_ws;
  const size_t nQK = (size_t)B_ * H_ * N_ * D_;   // 131072
  const size_t nV  = (size_t)B_ * N_ * HD_;       // 131072
  float* Qw = ws;
  float* Kw = Qw + nQK;
  float* Vw = Kw + nQK;
  float* qk = Vw + nV;                            // B*H*N*N = 4,194,304 floats

  // Stage A: projections — one block per (b,i) row
  proj_kernel<<<B_ * N_, IN_DIM_, 0, stream>>>(h, Wq, Wk, Wv, Qw, Kw, Vw);

  // Stage B: QK^T scores via f32 WMMA — one wave32 per 16x16 tile
  const int tiles = N_ / 16;
  qk_wmma_kernel<<<B_ * H_ * tiles * tiles, 32, 0, stream>>>(Qw, Kw, qk);

  // Stage C: fused online-softmax streaming pass (HBM-bound, ~512 MiB once)
  attn_stream_kernel<<<B_ * N_, 256, 0, stream>>>(e_att, e_val, mask, Vw, qk, out);
}